// NIQE_45775761441389
// MI455X (gfx1250) — compile-verified
//
#include <hip/hip_runtime.h>
#include <cmath>

// ---------------------------------------------------------------------------
// NIQE pipeline for gfx1250 (MI455X, wave32).
//   X: (4,1,1920,1920) f32   mu_pris: (4,36)   cov_pris: (4,36,36)
//   out: (4,) f32
// Stages: gauss7x7 MSCN (scale 1) -> block AGGD feats -> bicubic x0.5 resize
//         -> gauss7x7 MSCN (scale 2) -> block AGGD feats -> mean/center ->
//         cov via V_WMMA_F32_16X16X4_F32 -> SPD Cholesky solve -> sqrt.
// ---------------------------------------------------------------------------

typedef float v2f __attribute__((ext_vector_type(2)));
typedef float v8f __attribute__((ext_vector_type(8)));

struct GW { float w[49]; };

// exact dyadic bicubic(antialias, scale=0.5) weights; taps at 2*i + (-3..+4)
__constant__ float RW[8] = {-0.01171875f, -0.03515625f, 0.11328125f, 0.43359375f,
                             0.43359375f,  0.11328125f, -0.03515625f, -0.01171875f};

// --------------------------- R_GAM lookup table ----------------------------
__global__ void k_rgam(float* __restrict__ rgam) {
  int i = blockIdx.x * blockDim.x + threadIdx.x;
  if (i < 9801) {
    float g = (float)((double)i * 0.001 + 0.2);
    rgam[i] = expf(2.0f * lgammaf(2.0f / g) - lgammaf(1.0f / g) - lgammaf(3.0f / g));
  }
}

// ----------------- 7x7 Gaussian filter + MSCN normalization ----------------
// grid (W/16, H/16, B), block (16,16); edge-clamped halo of 3 in LDS.
__global__ void k_convnorm(const float* __restrict__ img, float* __restrict__ norm,
                           int H, int W, GW gw) {
  __shared__ float t[22][22];
  int b = blockIdx.z;
  const float* I = img + (size_t)b * H * W;
  int gx0 = blockIdx.x * 16 - 3, gy0 = blockIdx.y * 16 - 3;
  int tid = threadIdx.y * 16 + threadIdx.x;
  for (int idx = tid; idx < 22 * 22; idx += 256) {
    int iy = idx / 22, ix = idx - iy * 22;
    int y = gy0 + iy; y = y < 0 ? 0 : (y >= H ? H - 1 : y);
    int x = gx0 + ix; x = x < 0 ? 0 : (x >= W ? W - 1 : x);
    t[iy][ix] = I[(size_t)y * W + x];
  }
  __syncthreads();
  float mu = 0.f, m2 = 0.f;
#pragma unroll
  for (int dy = 0; dy < 7; dy++)
#pragma unroll
    for (int dx = 0; dx < 7; dx++) {
      float v = t[threadIdx.y + dy][threadIdx.x + dx];
      float w = gw.w[dy * 7 + dx];
      mu += w * v;
      m2 += w * v * v;
    }
  float c = t[threadIdx.y + 3][threadIdx.x + 3];
  float sg = sqrtf(fabsf(m2 - mu * mu));
  int ox = blockIdx.x * 16 + threadIdx.x, oy = blockIdx.y * 16 + threadIdx.y;
  norm[((size_t)b * H + oy) * W + ox] = (c - mu) / (sg + 1.0f);
}

// --------------------- bicubic x0.5: vertical pass (÷255) ------------------
__global__ void k_resizeV(const float* __restrict__ X, float* __restrict__ out,
                          int Hin, int W) {
  int b = blockIdx.z;
  int x  = blockIdx.x * 16 + threadIdx.x;
  int y2 = blockIdx.y * 16 + threadIdx.y;
  int Hout = Hin >> 1;
  if (x >= W || y2 >= Hout) return;
  const float* I = X + (size_t)b * Hin * W;
  float s = 0.f;
#pragma unroll
  for (int t = 0; t < 8; t++) {
    int yy = 2 * y2 - 3 + t;
    if (yy < 0) yy = -yy - 1; else if (yy >= Hin) yy = 2 * Hin - 1 - yy;
    s += RW[t] * (I[(size_t)yy * W + x] / 255.0f);
  }
  out[((size_t)b * Hout + y2) * W + x] = s;
}

// --------------------- bicubic x0.5: horizontal pass (×255) ----------------
__global__ void k_resizeH(const float* __restrict__ in, float* __restrict__ out,
                          int H, int Win) {
  int b = blockIdx.z;
  int x2 = blockIdx.x * 16 + threadIdx.x;
  int y  = blockIdx.y * 16 + threadIdx.y;
  int Wout = Win >> 1;
  if (x2 >= Wout || y >= H) return;
  const float* I = in + (size_t)b * H * Win;
  float s = 0.f;
#pragma unroll
  for (int t = 0; t < 8; t++) {
    int xx = 2 * x2 - 3 + t;
    if (xx < 0) xx = -xx - 1; else if (xx >= Win) xx = 2 * Win - 1 - xx;
    s += RW[t] * I[(size_t)y * Win + xx];
  }
  out[((size_t)b * H + y) * Wout + x2] = s * 255.0f;
}

// -------------------------- per-block AGGD features ------------------------
// grid (400, B), 256 threads. 5 signals: raw + 4 circular-shift products.
__global__ void k_blockfeats(const float* __restrict__ norm,
                             const float* __restrict__ rgam,
                             float* __restrict__ dp,
                             int bs, int H, int W, int featBase) {
  __shared__ float sb[9216];          // up to 96x96 block
  __shared__ float red[6][256];
  __shared__ float bestd[256];
  __shared__ int   besti[256];
  __shared__ float s_ls, s_rs, s_rn;

  int blk = blockIdx.x, b = blockIdx.y;
  int by = blk / 20, bx = blk - by * 20;
  int tid = threadIdx.x;
  int N = bs * bs;

  for (int idx = tid; idx < N; idx += 256) {
    int i = idx / bs, j = idx - i * bs;
    sb[idx] = norm[((size_t)b * H + by * bs + i) * W + bx * bs + j];
  }
  __syncthreads();

  const int SI[5] = {0, 0, 1, 1, 1};
  const int SJ[5] = {0, 1, 0, 1, -1};

  for (int sig = 0; sig < 5; sig++) {
    float ns = 0, nc = 0, ps = 0, pc = 0, ab = 0, sq = 0;
    int si = SI[sig], sj = SJ[sig];
    for (int idx = tid; idx < N; idx += 256) {
      float x = sb[idx];
      if (sig > 0) {
        int i = idx / bs, j = idx - i * bs;
        int i2 = i - si; if (i2 < 0) i2 += bs;
        int j2 = j - sj; if (j2 < 0) j2 += bs; else if (j2 >= bs) j2 -= bs;
        x *= sb[i2 * bs + j2];
      }
      float x2 = x * x;
      if (x < 0.f) { ns += x2; nc += 1.f; }
      if (x > 0.f) { ps += x2; pc += 1.f; }
      ab += fabsf(x);
      sq += x2;
    }
    red[0][tid] = ns; red[1][tid] = nc; red[2][tid] = ps;
    red[3][tid] = pc; red[4][tid] = ab; red[5][tid] = sq;
    __syncthreads();
    for (int off = 128; off > 0; off >>= 1) {
      if (tid < off)
        for (int k = 0; k < 6; k++) red[k][tid] += red[k][tid + off];
      __syncthreads();
    }
    if (tid == 0) {
      float ls = sqrtf(red[0][0] / red[1][0]);
      float rs = sqrtf(red[2][0] / red[3][0]);
      float gh = ls / rs;
      float m1 = red[4][0] / (float)N;
      float m2 = red[5][0] / (float)N;
      float rhat = m1 * m1 / m2;
      float g2 = gh * gh;
      float rn = rhat * (g2 * gh + 1.f) * (gh + 1.f) / ((g2 + 1.f) * (g2 + 1.f));
      s_ls = ls; s_rs = rs; s_rn = rn;
    }
    __syncthreads();
    float rn = s_rn;
    float bd = 3.4e38f; int bi = 0;
    for (int i = tid; i < 9801; i += 256) {
      float dd = rgam[i] - rn;
      float d2 = dd * dd;
      if (d2 < bd || (d2 == bd && i < bi)) { bd = d2; bi = i; }
    }
    bestd[tid] = bd; besti[tid] = bi;
    __syncthreads();
    for (int off = 128; off > 0; off >>= 1) {
      if (tid < off) {
        if (bestd[tid + off] < bestd[tid] ||
            (bestd[tid + off] == bestd[tid] && besti[tid + off] < besti[tid])) {
          bestd[tid] = bestd[tid + off];
          besti[tid] = besti[tid + off];
        }
      }
      __syncthreads();
    }
    if (tid == 0) {
      float alpha = (float)((double)besti[0] * 0.001 + 0.2);
      float ratio = expf(0.5f * (lgammaf(1.f / alpha) - lgammaf(3.f / alpha)));
      float bl = s_ls * ratio, br = s_rs * ratio;
      float* f = dp + ((size_t)(b * 400 + blk)) * 48 + featBase;
      if (sig == 0) {
        f[0] = alpha;
        f[1] = 0.5f * (bl + br);
      } else {
        float mean = (br - bl) * expf(lgammaf(2.f / alpha) - lgammaf(1.f / alpha));
        int base = 2 + 4 * (sig - 1);
        f[base] = alpha; f[base + 1] = mean; f[base + 2] = bl; f[base + 3] = br;
      }
    }
    __syncthreads();
  }
}

// ---------------- mean over blocks + centered (zero-padded) d --------------
__global__ void k_stats(const float* __restrict__ dp, float* __restrict__ mu_d,
                        float* __restrict__ dbuf) {
  int b = blockIdx.y, f = blockIdx.x, tid = threadIdx.x;
  __shared__ float r[128];
  __shared__ float smean;
  float s = 0.f;
  if (f < 36)
    for (int n = tid; n < 400; n += 128) s += dp[((size_t)(b * 400 + n)) * 48 + f];
  r[tid] = s;
  __syncthreads();
  for (int off = 64; off > 0; off >>= 1) {
    if (tid < off) r[tid] += r[tid + off];
    __syncthreads();
  }
  if (tid == 0) { smean = r[0] / 400.f; mu_d[b * 48 + f] = (f < 36) ? smean : 0.f; }
  __syncthreads();
  float mean = smean;
  for (int n = tid; n < 400; n += 128) {
    size_t o = ((size_t)(b * 400 + n)) * 48 + f;
    dbuf[o] = (f < 36) ? (dp[o] - mean) : 0.f;
  }
}

// ------------------ cov = d^T d / 399 via V_WMMA_F32_16X16X4 ----------------
// one wave32 per 16x16 output tile; grid (3,3,B).
// A 16x4 layout: lanes 0-15 hold K=0(v0)/K=1(v1); lanes 16-31 hold K=2/K=3.
// B 4x16 mirrors; C/D: vgpr r -> row r + (lane>=16 ? 8 : 0), col = lane%16.
__global__ void k_cov(const float* __restrict__ dbuf, float* __restrict__ cov) {
  int ti = blockIdx.x, tj = blockIdx.y, b = blockIdx.z;
  int lane = threadIdx.x;
  int half = lane >> 4;
  int l = lane & 15;
  const float* D = dbuf + (size_t)b * 400 * 48;
  int ca = ti * 16 + l;   // feature (row of cov tile)
  int cb = tj * 16 + l;   // feature (col of cov tile)
  v8f acc = {};
  for (int k = 0; k < 400; k += 4) {
    int k0 = k + half * 2;
    v2f a, bb;
    a[0]  = D[(size_t)(k0 + 0) * 48 + ca];
    a[1]  = D[(size_t)(k0 + 1) * 48 + ca];
    bb[0] = D[(size_t)(k0 + 0) * 48 + cb];
    bb[1] = D[(size_t)(k0 + 1) * 48 + cb];
    acc = __builtin_amdgcn_wmma_f32_16x16x4_f32(false, a, false, bb,
                                                (short)0, acc, false, false);
  }
  const float scl = 1.0f / 399.0f;
  int col = tj * 16 + l;
#pragma unroll
  for (int r = 0; r < 8; r++) {
    int row = ti * 16 + r + half * 8;
    if (row < 36 && col < 36)
      cov[((size_t)b * 36 + row) * 36 + col] = acc[r] * scl;
  }
}

// --------- E = (cov_pris+cov_dist)/2 SPD; out = sqrt(diff^T E^-1 diff) ------
__global__ void k_solve(const float* __restrict__ covp, const float* __restrict__ covd,
                        const float* __restrict__ mup, const float* __restrict__ mud,
                        float* __restrict__ out) {
  int b = blockIdx.x, tid = threadIdx.x;
  __shared__ float E[36][37];
  __shared__ float diff[36];
  for (int idx = tid; idx < 36 * 36; idx += 64) {
    int i = idx / 36, j = idx - i * 36;
    E[i][j] = 0.5f * (covp[(size_t)b * 1296 + idx] + covd[(size_t)b * 1296 + idx]);
  }
  for (int i = tid; i < 36; i += 64) diff[i] = mup[b * 36 + i] - mud[b * 48 + i];
  __syncthreads();
  for (int k = 0; k < 36; k++) {
    if (tid == 0) E[k][k] = sqrtf(E[k][k]);
    __syncthreads();
    for (int i = k + 1 + tid; i < 36; i += 64) E[i][k] /= E[k][k];
    __syncthreads();
    for (int i = k + 1 + tid; i < 36; i += 64) {
      float lik = E[i][k];
      for (int j = k + 1; j <= i; j++) E[i][j] -= lik * E[j][k];
    }
    __syncthreads();
  }
  if (tid == 0) {
    float w[36], acc = 0.f;
    for (int i = 0; i < 36; i++) {
      float s = diff[i];
      for (int j = 0; j < i; j++) s -= E[i][j] * w[j];
      w[i] = s / E[i][i];
      acc += w[i] * w[i];
    }
    out[b] = sqrtf(acc);
  }
}

// ---------------------------------------------------------------------------
extern "C" void kernel_launch(void* const* d_in, const int* in_sizes, int n_in,
                              void* d_out, int out_size, void* d_ws, size_t ws_size,
                              hipStream_t stream) {
  const float* X    = (const float*)d_in[0];   // (4,1,1920,1920)
  const float* mup  = (const float*)d_in[1];   // (4,36)
  const float* covp = (const float*)d_in[2];   // (4,36,36)
  float* out = (float*)d_out;                  // (4,)

  const int B = 4, H1 = 1920, W1 = 1920, H2 = 960, W2 = 960;

  // workspace layout (256B aligned)
  size_t off = 0;
  auto take = [&](size_t bytes) -> float* {
    float* p = (float*)((char*)d_ws + off);
    off += (bytes + 255) & ~(size_t)255;
    return p;
  };
  float* rgam  = take(9801 * sizeof(float));
  float* dp    = take((size_t)B * 400 * 48 * sizeof(float));
  float* mu_d  = take((size_t)B * 48 * sizeof(float));
  float* dbuf  = take((size_t)B * 400 * 48 * sizeof(float));
  float* cov_d = take((size_t)B * 36 * 36 * sizeof(float));
  float* regA  = take((size_t)B * H1 * W1 * sizeof(float));  // 59 MB, multi-use
  float* norm1 = regA;                                        // whole region
  float* tmp   = regA;                                        // after feats1
  float* img2  = regA + (size_t)B * H2 * W1;
  float* norm2 = img2 + (size_t)B * H2 * W2;
  (void)ws_size; (void)in_sizes; (void)n_in; (void)out_size;

  // Gaussian 7x7 window (float64 -> f32, matching reference)
  GW gw;
  {
    const double sig = 7.0 / 6.0;
    double g[49], s = 0.0;
    for (int y = 0; y < 7; y++)
      for (int x = 0; x < 7; x++) {
        double dy = y - 3, dx = x - 3;
        double v = std::exp(-(dx * dx + dy * dy) / (2.0 * sig * sig));
        g[y * 7 + x] = v; s += v;
      }
    for (int i = 0; i < 49; i++) gw.w[i] = (float)(g[i] / s);
  }

  // 1) Gamma-ratio lookup table
  k_rgam<<<(9801 + 255) / 256, 256, 0, stream>>>(rgam);
  // 2) scale-1 MSCN
  k_convnorm<<<dim3(W1 / 16, H1 / 16, B), dim3(16, 16), 0, stream>>>(X, norm1, H1, W1, gw);
  // 3) scale-1 block features (feat cols 0..17)
  k_blockfeats<<<dim3(400, B), 256, 0, stream>>>(norm1, rgam, dp, 96, H1, W1, 0);
  // 4) bicubic x0.5 (vertical then horizontal), /255 in, *255 out
  k_resizeV<<<dim3(W1 / 16, H2 / 16, B), dim3(16, 16), 0, stream>>>(X, tmp, H1, W1);
  k_resizeH<<<dim3(W2 / 16, H2 / 16, B), dim3(16, 16), 0, stream>>>(tmp, img2, H2, W1);
  // 5) scale-2 MSCN
  k_convnorm<<<dim3(W2 / 16, H2 / 16, B), dim3(16, 16), 0, stream>>>(img2, norm2, H2, W2, gw);
  // 6) scale-2 block features (feat cols 18..35)
  k_blockfeats<<<dim3(400, B), 256, 0, stream>>>(norm2, rgam, dp, 48, H2, W2, 18);
  // 7) per-feature mean + centered, zero-padded d
  k_stats<<<dim3(48, B), 128, 0, stream>>>(dp, mu_d, dbuf);
  // 8) covariance GEMM via f32 WMMA (one wave per 16x16 tile)
  k_cov<<<dim3(3, 3, B), 32, 0, stream>>>(dbuf, cov_d);
  // 9) SPD Cholesky solve + Mahalanobis sqrt
  k_solve<<<B, 64, 0, stream>>>(covp, cov_d, mup, mu_d, out);
}